// ViterbiDecoderForward_32203664786109
// MI455X (gfx1250) — compile-verified
//
#include <hip/hip_runtime.h>
#include <stdint.h>

// Viterbi decode + forward logsumexp for a linear-chain CRF.
// One workgroup per batch element; 256 threads (8 wave32).
// Compute mapping: tid = j*4 + part, j in [0,48) labels, part in [0,4) splits
// the 48-wide i-reduction into 12-element chunks, combined with wave32
// __shfl_xor butterflies (4-lane groups are wave-aligned).
// Emission tiles (48x48 f32) are double-buffered in LDS via CDNA5
// global_load_async_to_lds_b128 (ASYNCcnt) to overlap the next step's load
// with the current step's compute.

#define NL        48
#define START_LBL 46
#define END_LBL   47
#define TILE_STRIDE 52   // padded row stride (floats): 12*52 % 64 == 48 -> 4 parts hit
                         // distinct LDS banks; 52*4 bytes is 16B aligned for b128.
#define MAX_T     256

typedef int v4i __attribute__((vector_size(16)));
typedef __attribute__((address_space(1))) v4i glob_v4i;
typedef __attribute__((address_space(3))) v4i lds_v4i;
typedef __attribute__((address_space(3))) void lds_void;

__device__ __forceinline__ unsigned lds_byte_offset(const void* p) {
  return (unsigned)(uintptr_t)(lds_void*)p;
}

__device__ __forceinline__ void async_copy_f4(float* lds_dst, const float* gsrc) {
#if __has_builtin(__builtin_amdgcn_global_load_async_to_lds_b128)
  __builtin_amdgcn_global_load_async_to_lds_b128(
      (glob_v4i*)gsrc, (lds_v4i*)lds_dst, /*offset=*/0, /*cpol=*/0);
#else
  asm volatile("global_load_async_to_lds_b128 %0, %1, off"
               :: "v"(lds_byte_offset(lds_dst)), "v"(gsrc)
               : "memory");
#endif
}

__device__ __forceinline__ void wait_async0() {
#if __has_builtin(__builtin_amdgcn_s_wait_asynccnt)
  __builtin_amdgcn_s_wait_asynccnt(0);
#else
  asm volatile("s_wait_asynccnt 0" ::: "memory");
#endif
}

// Copy one 48x48 f32 tile (2304 floats = 576 float4) global -> padded LDS tile.
__device__ __forceinline__ void issue_tile_async(float* dst, const float* src, int tid) {
#pragma unroll
  for (int e0 = 0; e0 < 3; ++e0) {
    int e = tid + e0 * 256;
    if (e < (NL * NL / 4)) {
      int i = e / 12;           // row
      int c = e % 12;           // float4 column chunk
      async_copy_f4(dst + i * TILE_STRIDE + c * 4, src + (size_t)e * 4);
    }
  }
}

__global__ __launch_bounds__(256)
void ViterbiDecoderForward_kernel(const float* __restrict__ em,
                                  const float* __restrict__ trans,
                                  const int*   __restrict__ lengths,
                                  float*       __restrict__ out,
                                  int T, int B) {
  __shared__ __align__(16) float tileA[NL * TILE_STRIDE];
  __shared__ __align__(16) float tileB[NL * TILE_STRIDE];
  __shared__ float best_s[NL];
  __shared__ float upto_s[NL];
  __shared__ unsigned char bp_s[MAX_T * NL];

  const int b   = blockIdx.x;
  const int tid = threadIdx.x;
  const int len = lengths[b];
  const int part = tid & 3;
  const int j    = tid >> 2;
  const bool comp = (tid < NL * 4);      // 192 compute threads

  const float* emb = em + (size_t)b * T * (NL * NL);

  // Per-thread transition slice (constant over t): rows part*12..part*12+11, col j.
  float tr[12];
  if (comp) {
#pragma unroll
    for (int k = 0; k < 12; ++k) tr[k] = trans[(part * 12 + k) * NL + j];
  }

  // Prologue: async-stage tiles for t=0 and t=1.
  issue_tile_async(tileA, emb, tid);                     // t = 0
  issue_tile_async(tileB, emb + (size_t)(NL * NL), tid); // t = 1
  wait_async0();
  __syncthreads();

  // Init from row START of s_0, plus bp row 0.
  if (tid < NL) {
    const bool m0 = (len > 0);
    float v = m0 ? (tileA[START_LBL * TILE_STRIDE + tid] + trans[START_LBL * NL + tid])
                 : 0.0f;
    best_s[tid] = v;
    upto_s[tid] = v;
    bp_s[tid] = m0 ? (unsigned char)START_LBL : (unsigned char)END_LBL;
  }
  __syncthreads();

  float* cur = tileB;   // holds s_1
  float* nxt = tileA;   // free for s_2

  for (int t = 1; t < T; ++t) {
    const bool active = (t < len);                       // block-uniform
    const bool ldn    = (t + 1 < T) && ((t + 1) < len);  // block-uniform
    if (ldn) issue_tile_async(nxt, emb + (size_t)(t + 1) * NL * NL, tid);

    if (active) {
      float nb = 0.0f, nu = 0.0f;
      int bpj = END_LBL;
      if (comp) {
        float mb = -INFINITY, mu = -INFINITY;
        int bi = 0;
        float xu[12];
#pragma unroll
        for (int k = 0; k < 12; ++k) {
          const int i = part * 12 + k;
          const float s  = cur[i * TILE_STRIDE + j] + tr[k];
          const float cb = s + best_s[i];
          if (cb > mb) { mb = cb; bi = i; }
          const float cu = s + upto_s[i];
          xu[k] = cu;
          mu = fmaxf(mu, cu);
        }
        // Butterfly max/argmax + max over the 4 parts (lanes d apart, in-wave).
#pragma unroll
        for (int d = 1; d < 4; d <<= 1) {
          const float ov = __shfl_xor(mb, d, 32);
          const int   oi = __shfl_xor(bi, d, 32);
          if (ov > mb || (ov == mb && oi < bi)) { mb = ov; bi = oi; }
          mu = fmaxf(mu, __shfl_xor(mu, d, 32));
        }
        float es = 0.0f;
#pragma unroll
        for (int k = 0; k < 12; ++k) es += __expf(xu[k] - mu);
#pragma unroll
        for (int d = 1; d < 4; d <<= 1) es += __shfl_xor(es, d, 32);
        nb = mb;
        nu = mu + __logf(es);
        bpj = bi;
      }
      __syncthreads();   // all reads of best_s/upto_s done before update
      if (comp && part == 0) {
        best_s[j] = nb;
        upto_s[j] = nu;
        bp_s[t * NL + j] = (unsigned char)bpj;
      }
    } else {
      if (comp && part == 0) bp_s[t * NL + j] = (unsigned char)END_LBL;
    }

    if (ldn) wait_async0();
    __syncthreads();     // publish best/upto/bp + next tile ready
    float* tmp = cur; cur = nxt; nxt = tmp;
  }

  // Epilogue: score + backtrace (sequential, single thread per block).
  if (tid == 0) {
    out[(size_t)B * T + b] = -(upto_s[END_LBL] - best_s[END_LBL]);
    float* dec = out + (size_t)b * T;
    int ptr = END_LBL;
    for (int t = T - 1; t >= 1; --t) {
      const int d = (int)bp_s[t * NL + ptr];
      dec[t - 1] = (float)d;
      ptr = d;
    }
    dec[T - 1] = (float)END_LBL;
  }
}

extern "C" void kernel_launch(void* const* d_in, const int* in_sizes, int n_in,
                              void* d_out, int out_size, void* d_ws, size_t ws_size,
                              hipStream_t stream) {
  const float* em      = (const float*)d_in[0];
  const float* trans   = (const float*)d_in[1];
  const int*   lengths = (const int*)d_in[2];
  float*       out     = (float*)d_out;

  const int B = in_sizes[2];
  const int T = in_sizes[0] / (B * NL * NL);

  ViterbiDecoderForward_kernel<<<B, 256, 0, stream>>>(em, trans, lengths, out, T, B);
}